// FGN_layer_33406255628828
// MI455X (gfx1250) — compile-verified
//
#include <hip/hip_runtime.h>
#include <hip/hip_bf16.h>
#include <math.h>

// FGN layer:  res = (X W^T + b) * g,   g = exp(-sum_i w2*(x-c)^2),  w2 = min(ic,1e8)^2
// Expanded:   q = (X^2)(w2)^T - 2*X(w2 c)^T + t3,   t3[z] = sum_i w2*c^2
// => three f32 WMMA GEMMs (16x16x4) + elementwise epilogue.
// B=1024, IN=256, OUT=1024.

typedef __attribute__((ext_vector_type(2))) float v2f;
typedef __attribute__((ext_vector_type(8))) float v8f;

#define B_DIM   1024
#define IN_DIM  256
#define OUT_DIM 1024
#define INV_EPS 1.0e8f   // 1/EPS clamp from reference

static __device__ __forceinline__ v8f wmma_f32(v2f a, v2f b, v8f c) {
    return __builtin_amdgcn_wmma_f32_16x16x4_f32(false, a, false, b,
                                                 (short)0, c, false, false);
}

// ---------------------------------------------------------------------------
// Prep: materialize wc = w2*c, w2 = clamp(ic)^2, t3[z] = sum_i w2*c^2.
// One wave per row z; lane covers 8 elements (4 x float2, stride 64).
// ---------------------------------------------------------------------------
__global__ __launch_bounds__(256) void fgn_prep_kernel(
    const float* __restrict__ cen,   // [OUT, IN]
    const float* __restrict__ ic,    // [OUT, IN]
    float* __restrict__ wc,          // [OUT, IN]
    float* __restrict__ w2,          // [OUT, IN]
    float* __restrict__ t3)          // [OUT]
{
    const unsigned lane = threadIdx.x & 31;
    const unsigned wave = threadIdx.x >> 5;
    const unsigned z    = blockIdx.x * 8 + wave;

    const unsigned row = z * IN_DIM;
    float t3p = 0.0f;
#pragma unroll
    for (unsigned j = 0; j < 4; ++j) {
        const unsigned idx = row + j * 64 + lane * 2;
        v2f c = *(const v2f*)(cen + idx);
        v2f i = *(const v2f*)(ic  + idx);
        i.x = fminf(i.x, INV_EPS);
        i.y = fminf(i.y, INV_EPS);
        v2f s2 = i * i;          // w2
        v2f sc = s2 * c;         // w2*c
        *(v2f*)(w2 + idx) = s2;
        *(v2f*)(wc + idx) = sc;
        t3p += sc.x * c.x + sc.y * c.y;   // w2*c^2
    }
#pragma unroll
    for (int off = 16; off >= 1; off >>= 1)
        t3p += __shfl_xor(t3p, off, 32);
    if (lane == 0) t3[z] = t3p;
}

// ---------------------------------------------------------------------------
// Main: triple GEMM over precomputed operands, double-buffered.
// Wave = 32(M) x 32(N): 2x2 16-tiles -> 12 WMMAs per 8 b64 loads per k-step.
// Block = 8 waves = 2(Mw) x 4(Nw) = 64(M) x 128(N).
// ---------------------------------------------------------------------------
struct Frags {
    v2f a[2];                 // A fragments for 2 M-tiles
    v2f bw[2], bwc[2], bw2[2]; // B fragments for 2 N-tiles x 3 matrices
};

static __device__ __forceinline__ Frags load_frags(
    const float* __restrict__ x, const float* __restrict__ w,
    const float* __restrict__ wc, const float* __restrict__ w2,
    unsigned idxA, unsigned idxN, unsigned k)
{
    Frags f;
#pragma unroll
    for (unsigned mt = 0; mt < 2; ++mt)
        f.a[mt] = *(const v2f*)(x + (idxA + mt * 16u * IN_DIM + k));
#pragma unroll
    for (unsigned nt = 0; nt < 2; ++nt) {
        const unsigned off = idxN + nt * 16u * IN_DIM + k;
        f.bw[nt]  = *(const v2f*)(w  + off);
        f.bwc[nt] = *(const v2f*)(wc + off);
        f.bw2[nt] = *(const v2f*)(w2 + off);
    }
    return f;
}

static __device__ __forceinline__ void compute_frags(
    const Frags& f, v8f acc_l[2][2], v8f acc_t2[2][2], v8f acc_t1[2][2])
{
    v2f a2[2];
#pragma unroll
    for (unsigned mt = 0; mt < 2; ++mt) a2[mt] = f.a[mt] * f.a[mt];
#pragma unroll
    for (unsigned mt = 0; mt < 2; ++mt) {
#pragma unroll
        for (unsigned nt = 0; nt < 2; ++nt) {
            acc_l[mt][nt]  = wmma_f32(f.a[mt], f.bw[nt],  acc_l[mt][nt]);
            acc_t2[mt][nt] = wmma_f32(f.a[mt], f.bwc[nt], acc_t2[mt][nt]);
            acc_t1[mt][nt] = wmma_f32(a2[mt],  f.bw2[nt], acc_t1[mt][nt]);
        }
    }
}

__global__ __launch_bounds__(256) void fgn_main_kernel(
    const float* __restrict__ x,     // [B, IN]
    const float* __restrict__ w,     // [OUT, IN]
    const float* __restrict__ bias,  // [OUT]
    const float* __restrict__ wc,    // [OUT, IN]
    const float* __restrict__ w2,    // [OUT, IN]
    const float* __restrict__ t3,    // [OUT]
    float* __restrict__ out_res,     // [B, OUT]
    float* __restrict__ out_g)       // [B, OUT]
{
    const unsigned lane  = threadIdx.x & 31;
    const unsigned wave  = threadIdx.x >> 5;
    const unsigned laneM = lane & 15;
    const unsigned halfK = lane >> 4;

    const unsigned m0 = blockIdx.y * 64 + (wave & 1) * 32;
    const unsigned n0 = blockIdx.x * 128 + (wave >> 1) * 32;

    // 32-bit unsigned element indices (matrices are 256K elements).
    const unsigned idxA = (m0 + laneM) * IN_DIM + 2 * halfK;
    const unsigned idxN = (n0 + laneM) * IN_DIM + 2 * halfK;

    v8f acc_l[2][2], acc_t1[2][2], acc_t2[2][2];
#pragma unroll
    for (unsigned mt = 0; mt < 2; ++mt)
#pragma unroll
        for (unsigned nt = 0; nt < 2; ++nt) {
            acc_l[mt][nt]  = (v8f)0.0f;
            acc_t1[mt][nt] = (v8f)0.0f;
            acc_t2[mt][nt] = (v8f)0.0f;
        }

    // Software pipeline: stage k+4 loads behind stage k WMMAs.
    Frags cur = load_frags(x, w, wc, w2, idxA, idxN, 0);
    for (unsigned k = 0; k < IN_DIM - 8; k += 8) {
        Frags nxt = load_frags(x, w, wc, w2, idxA, idxN, k + 4);
        compute_frags(cur, acc_l, acc_t2, acc_t1);
        cur = load_frags(x, w, wc, w2, idxA, idxN, k + 8);
        compute_frags(nxt, acc_l, acc_t2, acc_t1);
    }
    {
        Frags nxt = load_frags(x, w, wc, w2, idxA, idxN, IN_DIM - 4);
        compute_frags(cur, acc_l, acc_t2, acc_t1);
        compute_frags(nxt, acc_l, acc_t2, acc_t1);
    }

#pragma unroll
    for (unsigned mt = 0; mt < 2; ++mt) {
#pragma unroll
        for (unsigned nt = 0; nt < 2; ++nt) {
            const unsigned n = n0 + nt * 16 + laneM;
            const float t3n = t3[n];
            const float bn  = bias[n];
#pragma unroll
            for (unsigned r = 0; r < 8; ++r) {
                const unsigned m = m0 + mt * 16 + r + 8 * halfK;
                const float l = acc_l[mt][nt][r] + bn;
                const float q = acc_t1[mt][nt][r] - 2.0f * acc_t2[mt][nt][r] + t3n;
                const float g = __expf(-q);
                const unsigned idx = m * OUT_DIM + n;
                out_res[idx] = l * g;
                out_g[idx]   = g;
            }
        }
    }
}

// ---------------------------------------------------------------------------
// Fallback: fully fused (no workspace), identical math done in-loop.
// Wave = 16(M) x 64(N), block = 8 waves = 128(M) x 64(N).
// ---------------------------------------------------------------------------
__global__ __launch_bounds__(256) void fgn_fused_kernel(
    const float* __restrict__ x, const float* __restrict__ w,
    const float* __restrict__ bias, const float* __restrict__ cen,
    const float* __restrict__ ic,
    float* __restrict__ out_res, float* __restrict__ out_g)
{
    const unsigned lane  = threadIdx.x & 31;
    const unsigned wave  = threadIdx.x >> 5;
    const unsigned laneM = lane & 15;
    const unsigned halfK = lane >> 4;

    const unsigned m0 = blockIdx.y * 128 + wave * 16;
    const unsigned n0 = blockIdx.x * 64;

    const unsigned idxA = (m0 + laneM) * IN_DIM + 2 * halfK;
    const unsigned idxN = (n0 + laneM) * IN_DIM + 2 * halfK;

    v8f acc_l[4], acc_t1[4], acc_t2[4];
    float t3p[4];
#pragma unroll
    for (unsigned t = 0; t < 4; ++t) {
        acc_l[t]  = (v8f)0.0f;
        acc_t1[t] = (v8f)0.0f;
        acc_t2[t] = (v8f)0.0f;
        t3p[t]    = 0.0f;
    }

    for (unsigned k = 0; k < IN_DIM; k += 4) {
        v2f a  = *(const v2f*)(x + idxA + k);
        v2f a2 = a * a;
#pragma unroll
        for (unsigned t = 0; t < 4; ++t) {
            const unsigned off = idxN + t * 16u * IN_DIM + k;
            v2f bw = *(const v2f*)(w   + off);
            v2f bc = *(const v2f*)(cen + off);
            v2f bi = *(const v2f*)(ic  + off);
            bi.x = fminf(bi.x, INV_EPS);
            bi.y = fminf(bi.y, INV_EPS);
            v2f s2 = bi * bi;
            v2f sc = s2 * bc;
            t3p[t] += sc.x * bc.x + sc.y * bc.y;
            acc_l[t]  = wmma_f32(a,  bw, acc_l[t]);
            acc_t2[t] = wmma_f32(a,  sc, acc_t2[t]);
            acc_t1[t] = wmma_f32(a2, s2, acc_t1[t]);
        }
    }

#pragma unroll
    for (unsigned t = 0; t < 4; ++t) {
        float t3full = t3p[t] + __shfl_xor(t3p[t], 16, 32);
        const unsigned n = n0 + t * 16 + laneM;
        const float bn = bias[n];
#pragma unroll
        for (unsigned r = 0; r < 8; ++r) {
            const unsigned m = m0 + r + 8 * halfK;
            const float l = acc_l[t][r] + bn;
            const float q = acc_t1[t][r] - 2.0f * acc_t2[t][r] + t3full;
            const float g = __expf(-q);
            const unsigned idx = m * OUT_DIM + n;
            out_res[idx] = l * g;
            out_g[idx]   = g;
        }
    }
}

extern "C" void kernel_launch(void* const* d_in, const int* in_sizes, int n_in,
                              void* d_out, int out_size, void* d_ws, size_t ws_size,
                              hipStream_t stream) {
    (void)in_sizes; (void)n_in; (void)out_size;
    const float* x    = (const float*)d_in[0];  // inputs     [1024, 256]
    const float* w    = (const float*)d_in[1];  // weights    [1024, 256]
    const float* bias = (const float*)d_in[2];  // biases     [1024]
    const float* cen  = (const float*)d_in[3];  // centers    [1024, 256]
    const float* ic   = (const float*)d_in[4];  // inv_covars [1024, 256]

    float* out_res = (float*)d_out;
    float* out_g   = (float*)d_out + (size_t)B_DIM * OUT_DIM;

    const size_t mat_elems = (size_t)OUT_DIM * IN_DIM;                   // 256K
    const size_t needed    = (2 * mat_elems + OUT_DIM) * sizeof(float);  // ~2.1 MB

    dim3 block(256);                       // 8 waves

    if (ws_size >= needed) {
        float* ws_wc = (float*)d_ws;
        float* ws_w2 = ws_wc + mat_elems;
        float* ws_t3 = ws_w2 + mat_elems;
        fgn_prep_kernel<<<dim3(OUT_DIM / 8), block, 0, stream>>>(cen, ic, ws_wc, ws_w2, ws_t3);
        // Main: block tile 64(M) x 128(N) -> grid 8(N) x 16(M)
        fgn_main_kernel<<<dim3(OUT_DIM / 128, B_DIM / 64), block, 0, stream>>>(
            x, w, bias, ws_wc, ws_w2, ws_t3, out_res, out_g);
    } else {
        fgn_fused_kernel<<<dim3(OUT_DIM / 64, B_DIM / 128), block, 0, stream>>>(
            x, w, bias, cen, ic, out_res, out_g);
    }
}